// TrajectoryPredictor_40029095199357
// MI455X (gfx1250) — compile-verified
//
#include <hip/hip_runtime.h>
#include <math.h>

// ---------------- problem constants (fixed by the reference harness) -------
#define Bv     4096
#define Tv     100
#define INv    5
#define Hv     20
#define ENCv   40
#define ATTNv  32
#define OUTv   5
#define TOUTv  30
#define BT     16      // batch tile per workgroup
#define NTHR   256     // 8 waves: waves 0-3 = dir0, waves 4-7 = dir1 (encoder)

typedef float v2f __attribute__((ext_vector_type(2)));
typedef float v8f __attribute__((ext_vector_type(8)));

__device__ __forceinline__ float fast_rcp(float x) { return __builtin_amdgcn_rcpf(x); }
__device__ __forceinline__ float sigm(float x) { return fast_rcp(1.0f + __expf(-x)); }
__device__ __forceinline__ float ftanh(float x)
{
#if __has_builtin(__builtin_amdgcn_tanhf)
    return __builtin_amdgcn_tanhf(x);      // CDNA5 V_TANH_F32
#else
    return tanhf(x);
#endif
}

// ---------------------------------------------------------------------------
// 16x64 = (16xKpad) @ W^T (+bias), f32 WMMA 16x16x4, guard-free:
//   A    : LDS, row-major [16][lda]  (rows = batch elements, K zero-padded)
//   W    : LDS, row-major [64][ldk]  (rows 60..63 zero-padded -> no selects)
//   bias : LDS, 64 entries (60..63 zero)
//   outB : LDS, row-major [16][64]   (cols 60..63 written but never read)
// One wave per 16-column tile; EXEC all-ones inside participating waves.
// A-frag (16x4 f32):  lane = 16*(k>=2) + m, elem j -> K = 2*(lane>>4)+j
// B-frag (4x16 f32):  elem j holds row K = j + 2*(lane>>4), col = lane&15
// C/D   (16x16 f32):  acc[r] = C[m = r + 8*(lane>>4)][n = lane&15]
// ---------------------------------------------------------------------------
__device__ __forceinline__ void gemm16(const float* A, int lda,
                                       const float* W, int ldk, int Kpad,
                                       const float* bias,
                                       float* outB, int wv, int lane)
{
    if (wv >= 4) return;
    int ncol = (wv << 4) + (lane & 15);
    int hi   = lane >> 4;
    float bval = bias[ncol];
    v8f acc;
#pragma unroll
    for (int r = 0; r < 8; ++r) acc[r] = bval;
    const float* arow = A + (lane & 15) * lda;
    const float* wr   = W + ncol * ldk;
    for (int kc = 0; kc < Kpad; kc += 4) {
        int k0 = kc + (hi << 1);
        v2f a, b;
        a.x = arow[k0];
        a.y = arow[k0 + 1];
        b.x = wr[k0];
        b.y = wr[k0 + 1];
        acc = __builtin_amdgcn_wmma_f32_16x16x4_f32(false, a, false, b,
                                                    (short)0, acc, false, false);
    }
#pragma unroll
    for (int r = 0; r < 8; ++r)
        outB[(r + (hi << 3)) * 64 + ncol] = acc[r];
}

__device__ __forceinline__ void cpy(float* d, const float* s, int n, int tid)
{
    for (int i = tid; i < n; i += NTHR) d[i] = s[i];
}
// padded matrix copy: dst [R][ldk] <- src [N][K], zero-filled outside
__device__ __forceinline__ void cpyW(float* d, const float* s, int N, int K,
                                     int R, int ldk, int tid)
{
    for (int i = tid; i < R * ldk; i += NTHR) {
        int r = i / ldk, k = i % ldk;
        d[i] = (r < N && k < K) ? s[r * K + k] : 0.0f;
    }
}
// padded bias copy: dst[R] <- src[N], zero-filled
__device__ __forceinline__ void cpyB(float* d, const float* s, int N, int R, int tid)
{
    for (int i = tid; i < R; i += NTHR) d[i] = (i < N) ? s[i] : 0.0f;
}

// ---------------------------------------------------------------------------
// Fully fused seq2seq: 2-layer biGRU encoder + attention + 2-layer GRU decoder.
// One block = 16 batch elements; per-step state lives in LDS; fwd/bwd GRU
// directions run concurrently on wave-halves of the block.
// ws layout: l0 [B][T][ENC] (b-major; later overlaid by enc_proj, stride T*ENC)
//            eo [B][T][ENC] at offset B*T*ENC      (needs 131,072,000 B total)
// ---------------------------------------------------------------------------
__global__ __launch_bounds__(NTHR) void traj_fused(
    const float* __restrict__ x,
    const float* __restrict__ eWih0, const float* __restrict__ eWhh0,
    const float* __restrict__ ebih0, const float* __restrict__ ebhh0,
    const float* __restrict__ eWih1, const float* __restrict__ eWhh1,
    const float* __restrict__ ebih1, const float* __restrict__ ebhh1,
    const float* __restrict__ dWih0g, const float* __restrict__ dWhh0g,
    const float* __restrict__ dbih0g, const float* __restrict__ dbhh0g,
    const float* __restrict__ dWih1g, const float* __restrict__ dWhh1g,
    const float* __restrict__ dbih1g, const float* __restrict__ dbhh1g,
    const float* __restrict__ Wh2d,  const float* __restrict__ bh2d,
    const float* __restrict__ Wattn, const float* __restrict__ battn,
    const float* __restrict__ vattn,
    const float* __restrict__ Wout,  const float* __restrict__ bout,
    float* __restrict__ out, float* __restrict__ ws)
{
    // ---- LDS pool (60.4 KB), phase-overlaid -------------------------------
    __shared__ float sm[15104];
    float* hB  = sm + 8192;    // [2][16*20] enc hidden / dec hid0,hid1
    float* hT  = sm + 8832;    // [2][16*20] enc finals   (overlay: satt [16*32])
    float* giB = sm + 9472;    // gi/gh: dir d at 9472 + d*2048 (gi then gh)
    float* xiB = sm + 13568;   // [2][16*48] step inputs  (overlay dir1: ybuf)

    const int tid  = threadIdx.x;
    const int wv   = tid >> 5, lane = tid & 31;
    const int ht   = tid & 127, dh = tid >> 7;   // half-block = one direction
    const int b0   = blockIdx.x * BT;

    float* l0 = ws;                               // layer-0 out / enc_proj
    float* eo = ws + (size_t)Bv * Tv * ENCv;      // enc_out [B][T][ENC]

    // ================= encoder layer 0 (both dirs concurrently) ============
    cpy(sm, eWih0, 2 * 60 * INv, tid);                       // [2][60][5] scalar use
    cpyW(sm + 600,  eWhh0,        60, Hv, 64, Hv, tid);      // dir0 [64][20]
    cpyW(sm + 1880, eWhh0 + 1200, 60, Hv, 64, Hv, tid);      // dir1
    cpy(sm + 3160, ebih0, 120, tid);                         // scalar use
    cpyB(sm + 3280, ebhh0,      60, 64, tid);
    cpyB(sm + 3344, ebhh0 + 60, 60, 64, tid);
    for (int i = tid; i < 2 * BT * Hv; i += NTHR) hB[i] = 0.0f;
    __syncthreads();
    {
        float* h  = hB  + dh * (BT * Hv);
        float* gi = giB + dh * 2048;
        float* gh = gi + 1024;
        float* xi = xiB + dh * (BT * 48);
        for (int t = 0; t < Tv; ++t) {
            int tt = dh ? (Tv - 1 - t) : t;
            for (int i = ht; i < BT * INv; i += 128) {
                int b = i / INv, c = i % INv;
                xi[b * 48 + c] = x[((size_t)(b0 + b) * Tv + tt) * INv + c];
            }
            __syncthreads();
            for (int i = ht; i < BT * 60; i += 128) {          // gi: K=5 scalar
                int b = i / 60, n = i % 60;
                const float* wr = sm + (dh * 60 + n) * INv;
                float s = sm[3160 + dh * 60 + n];
#pragma unroll
                for (int c = 0; c < INv; ++c) s += xi[b * 48 + c] * wr[c];
                gi[b * 64 + n] = s;
            }
            gemm16(h, Hv, sm + 600 + dh * 1280, Hv, Hv,        // gh: WMMA K=20
                   sm + 3280 + dh * 64, gh, wv & 3, lane);
            __syncthreads();
            for (int i = ht; i < BT * Hv; i += 128) {          // gates
                int b = i / Hv, j = i % Hv;
                float r  = sigm(gi[b * 64 + j]          + gh[b * 64 + j]);
                float z  = sigm(gi[b * 64 + j + Hv]     + gh[b * 64 + j + Hv]);
                float nn = ftanh(gi[b * 64 + j + 2*Hv] + r * gh[b * 64 + j + 2*Hv]);
                float hn = (1.0f - z) * nn + z * h[b * Hv + j];
                h[b * Hv + j] = hn;
                l0[((size_t)(b0 + b) * Tv + tt) * ENCv + dh * Hv + j] = hn;
            }
            __syncthreads();
        }
    }

    // ================= encoder layer 1 =====================================
    cpyW(sm,        eWih1,        60, ENCv, 64, ENCv, tid);  // dir0 [64][40]
    cpyW(sm + 2560, eWih1 + 2400, 60, ENCv, 64, ENCv, tid);
    cpyW(sm + 5120, eWhh1,        60, Hv, 64, Hv, tid);
    cpyW(sm + 6400, eWhh1 + 1200, 60, Hv, 64, Hv, tid);
    cpyB(sm + 7680, ebih1,      60, 64, tid);
    cpyB(sm + 7744, ebih1 + 60, 60, 64, tid);
    cpyB(sm + 7808, ebhh1,      60, 64, tid);
    cpyB(sm + 7872, ebhh1 + 60, 60, 64, tid);
    for (int i = tid; i < 2 * BT * Hv; i += NTHR) hB[i] = 0.0f;
    __syncthreads();
    {
        float* h  = hB  + dh * (BT * Hv);
        float* gi = giB + dh * 2048;
        float* gh = gi + 1024;
        float* xi = xiB + dh * (BT * 48);
        for (int t = 0; t < Tv; ++t) {
            int tt = dh ? (Tv - 1 - t) : t;
            for (int i = ht; i < BT * ENCv; i += 128) {
                int b = i / ENCv, c = i % ENCv;
                xi[b * 48 + c] = l0[((size_t)(b0 + b) * Tv + tt) * ENCv + c];
            }
            __syncthreads();
            gemm16(xi, 48, sm + dh * 2560, ENCv, ENCv,         // gi: WMMA K=40
                   sm + 7680 + dh * 64, gi, wv & 3, lane);
            gemm16(h, Hv, sm + 5120 + dh * 1280, Hv, Hv,       // gh: WMMA K=20
                   sm + 7808 + dh * 64, gh, wv & 3, lane);
            __syncthreads();
            for (int i = ht; i < BT * Hv; i += 128) {
                int b = i / Hv, j = i % Hv;
                float r  = sigm(gi[b * 64 + j]          + gh[b * 64 + j]);
                float z  = sigm(gi[b * 64 + j + Hv]     + gh[b * 64 + j + Hv]);
                float nn = ftanh(gi[b * 64 + j + 2*Hv] + r * gh[b * 64 + j + 2*Hv]);
                float hn = (1.0f - z) * nn + z * h[b * Hv + j];
                h[b * Hv + j] = hn;
                eo[((size_t)(b0 + b) * Tv + tt) * ENCv + dh * Hv + j] = hn;
            }
            __syncthreads();
        }
        for (int i = ht; i < BT * Hv; i += 128) hT[dh * (BT * Hv) + i] = h[i];
        __syncthreads();
    }

    // ============ h_dec, y_prev0, enc_proj (overlaid onto l0) ==============
    cpy(sm,       Wh2d, Hv * ENCv, tid);          // [20][40]
    cpy(sm + 800, bh2d, Hv, tid);
    for (int i = tid; i < ATTNv * ENCv; i += NTHR) {   // W_a_e = W_attn[:,20:]
        int a = i / ENCv, c = i % ENCv;
        sm[820 + i] = Wattn[a * (Hv + ENCv) + Hv + c];
    }
    cpy(sm + 2100, battn, ATTNv, tid);
    __syncthreads();

    float* hid0 = hB;
    float* hid1 = hB + BT * Hv;
    for (int i = tid; i < BT * Hv; i += NTHR) {        // h_dec = tanh([hT0,hT1].W_h2d^T + b)
        int b = i / Hv, j = i % Hv;
        float s = sm[800 + j];
        const float* wr = sm + j * ENCv;
#pragma unroll
        for (int c = 0; c < ENCv; ++c) {
            float e = (c < Hv) ? hT[b * Hv + c] : hT[BT * Hv + b * Hv + (c - Hv)];
            s += e * wr[c];
        }
        s = ftanh(s);
        hid0[i] = s; hid1[i] = s;
    }
    float* xi = xiB;                                   // decoder input [yp(5), w(40), 0(3)]
    for (int i = tid; i < BT * OUTv; i += NTHR) {      // y_prev0 = x[:, -1, :5]
        int b = i / OUTv, o = i % OUTv;
        xi[b * 48 + o] = x[((size_t)(b0 + b) * Tv + (Tv - 1)) * INv + o];
    }
    for (int i = tid; i < BT * 3; i += NTHR) xi[(i / 3) * 48 + 45 + (i % 3)] = 0.0f;
    for (int i = tid; i < BT * Tv * ATTNv; i += NTHR) {    // enc_proj
        int b = i / (Tv * ATTNv), r = i % (Tv * ATTNv), t = r / ATTNv, a = r % ATTNv;
        float s = sm[2100 + a];
        const float* wr = sm + 820 + a * ENCv;
        const float* er = eo + ((size_t)(b0 + b) * Tv + t) * ENCv;
        for (int c = 0; c < ENCv; ++c) s += er[c] * wr[c];
        l0[(size_t)(b0 + b) * Tv * ENCv + t * ATTNv + a] = s;
    }
    __syncthreads();

    // ================= decoder weights =====================================
    cpyW(sm,        dWih0g, 60, 45, 64, 48, tid);      // [64][48] zero-padded
    cpyW(sm + 3072, dWhh0g, 60, Hv, 64, Hv, tid);
    cpyW(sm + 4352, dWih1g, 60, Hv, 64, Hv, tid);
    cpyW(sm + 5632, dWhh1g, 60, Hv, 64, Hv, tid);
    cpyB(sm + 6912, dbih0g, 60, 64, tid);
    cpyB(sm + 6976, dbhh0g, 60, 64, tid);
    cpyB(sm + 7040, dbih1g, 60, 64, tid);
    cpyB(sm + 7104, dbhh1g, 60, 64, tid);
    for (int i = tid; i < ATTNv * Hv; i += NTHR) {     // W_a_h = W_attn[:,:20]
        int a = i / Hv, j = i % Hv;
        sm[7168 + i] = Wattn[a * (Hv + ENCv) + j];
    }
    cpy(sm + 7808, vattn, ATTNv, tid);
    cpy(sm + 7840, Wout, OUTv * (Hv + OUTv + ENCv), tid);
    cpy(sm + 8168, bout, OUTv, tid);
    __syncthreads();

    // ================= decoder loop ========================================
    float* satt = hT;                // [16][32] (hT is dead)
    float* sc   = giB + 2048;        // [16][100] (dir-1 gi/gh region is dead)
    float* gi   = giB;
    float* gh   = giB + 1024;
    float* ybuf = xiB + BT * 48;     // [16][8] (dir-1 xi region is dead)

    for (int st = 0; st < TOUTv; ++st) {
        for (int i = tid; i < BT * ATTNv; i += NTHR) {     // s_att = hd @ W_a_h^T
            int b = i / ATTNv, a = i % ATTNv;
            const float* wr = sm + 7168 + a * Hv;
            float s = 0.0f;
#pragma unroll
            for (int j = 0; j < Hv; ++j) s += hid1[b * Hv + j] * wr[j];
            satt[i] = s;
        }
        __syncthreads();
        for (int i = tid; i < BT * Tv; i += NTHR) {        // scores = tanh(proj+s).v
            int b = i / Tv, t = i % Tv;
            const float* pr = l0 + (size_t)(b0 + b) * Tv * ENCv + t * ATTNv;
            float s = 0.0f;
            for (int a = 0; a < ATTNv; ++a)
                s += ftanh(pr[a] + satt[b * ATTNv + a]) * sm[7808 + a];
            sc[b * Tv + t] = s;
        }
        __syncthreads();
        if (tid < BT) {                                    // softmax over T
            int b = tid;
            float m = -1e30f;
            for (int t = 0; t < Tv; ++t) m = fmaxf(m, sc[b * Tv + t]);
            float su = 0.0f;
            for (int t = 0; t < Tv; ++t) { float e = __expf(sc[b * Tv + t] - m); sc[b * Tv + t] = e; su += e; }
            float inv = fast_rcp(su);
            for (int t = 0; t < Tv; ++t) sc[b * Tv + t] *= inv;
        }
        __syncthreads();
        for (int i = tid; i < BT * ENCv; i += NTHR) {      // w = a @ enc_out
            int b = i / ENCv, c = i % ENCv;
            const float* er = eo + (size_t)(b0 + b) * Tv * ENCv + c;
            float s = 0.0f;
            for (int t = 0; t < Tv; ++t) s += sc[b * Tv + t] * er[(size_t)t * ENCv];
            xi[b * 48 + 5 + c] = s;
        }
        __syncthreads();
        // GRU cell 0: inp(45) -> h0
        gemm16(xi, 48, sm, 48, 48, sm + 6912, gi, wv, lane);
        gemm16(hid0, Hv, sm + 3072, Hv, Hv, sm + 6976, gh, wv, lane);
        __syncthreads();
        for (int i = tid; i < BT * Hv; i += NTHR) {
            int b = i / Hv, j = i % Hv;
            float r  = sigm(gi[b * 64 + j]          + gh[b * 64 + j]);
            float z  = sigm(gi[b * 64 + j + Hv]     + gh[b * 64 + j + Hv]);
            float nn = ftanh(gi[b * 64 + j + 2*Hv] + r * gh[b * 64 + j + 2*Hv]);
            hid0[i] = (1.0f - z) * nn + z * hid0[i];
        }
        __syncthreads();
        // GRU cell 1: h0 -> h1
        gemm16(hid0, Hv, sm + 4352, Hv, Hv, sm + 7040, gi, wv, lane);
        gemm16(hid1, Hv, sm + 5632, Hv, Hv, sm + 7104, gh, wv, lane);
        __syncthreads();
        for (int i = tid; i < BT * Hv; i += NTHR) {
            int b = i / Hv, j = i % Hv;
            float r  = sigm(gi[b * 64 + j]          + gh[b * 64 + j]);
            float z  = sigm(gi[b * 64 + j + Hv]     + gh[b * 64 + j + Hv]);
            float nn = ftanh(gi[b * 64 + j + 2*Hv] + r * gh[b * 64 + j + 2*Hv]);
            hid1[i] = (1.0f - z) * nn + z * hid1[i];
        }
        __syncthreads();
        for (int i = tid; i < BT * OUTv; i += NTHR) {      // y = [h1,yp,w] @ W_out^T + b
            int b = i / OUTv, o = i % OUTv;
            const float* wr = sm + 7840 + o * (Hv + OUTv + ENCv);
            float s = sm[8168 + o];
#pragma unroll
            for (int j = 0; j < Hv; ++j)  s += hid1[b * Hv + j] * wr[j];
#pragma unroll
            for (int o2 = 0; o2 < OUTv; ++o2) s += xi[b * 48 + o2] * wr[Hv + o2];
#pragma unroll
            for (int c = 0; c < ENCv; ++c) s += xi[b * 48 + 5 + c] * wr[Hv + OUTv + c];
            ybuf[b * 8 + o] = s;
            out[((size_t)(b0 + b) * TOUTv + st) * OUTv + o] = s;
        }
        __syncthreads();
        for (int i = tid; i < BT * OUTv; i += NTHR) {      // yp <- y
            int b = i / OUTv, o = i % OUTv;
            xi[b * 48 + o] = ybuf[b * 8 + o];
        }
        __syncthreads();
    }
}

// ---------------------------------------------------------------------------
// d_ws requirement: 2 * B*T*ENC * 4 bytes = 131,072,000 bytes
//   [0, B*T*ENC)         : layer-0 output, later overlaid by enc_proj
//   [B*T*ENC, 2*B*T*ENC) : enc_out
// ---------------------------------------------------------------------------
extern "C" void kernel_launch(void* const* d_in, const int* in_sizes, int n_in,
                              void* d_out, int out_size, void* d_ws, size_t ws_size,
                              hipStream_t stream)
{
    (void)in_sizes; (void)n_in; (void)out_size; (void)ws_size;
    traj_fused<<<dim3(Bv / BT), dim3(NTHR), 0, stream>>>(
        (const float*)d_in[0],
        (const float*)d_in[1],  (const float*)d_in[2],
        (const float*)d_in[3],  (const float*)d_in[4],
        (const float*)d_in[5],  (const float*)d_in[6],
        (const float*)d_in[7],  (const float*)d_in[8],
        (const float*)d_in[9],  (const float*)d_in[10],
        (const float*)d_in[11], (const float*)d_in[12],
        (const float*)d_in[13], (const float*)d_in[14],
        (const float*)d_in[15], (const float*)d_in[16],
        (const float*)d_in[17], (const float*)d_in[18],
        (const float*)d_in[19], (const float*)d_in[20],
        (const float*)d_in[21],
        (const float*)d_in[22], (const float*)d_in[23],
        (float*)d_out, (float*)d_ws);
}